// Model_61186104099220
// MI455X (gfx1250) — compile-verified
//
#include <hip/hip_runtime.h>
#include <hip/hip_bf16.h>

typedef __attribute__((ext_vector_type(2))) float v2f;
typedef __attribute__((ext_vector_type(8))) float v8f;

#define B_TOTAL   65536
#define SEQ_L     512
#define PRED_T    96
#define KDIM      100      // 3 lags + 96 exog + 1 ones-column
#define KCHUNKS   25       // KDIM / 4
#define VSTRIDE   102      // even stride (8B-aligned b64 LDS loads), != mult of 64
#define ROWS_PER_BLOCK 64  // 4 waves x 16 rows

// ---------------------------------------------------------------------------
// Kernel 1: build the 100 x 96 unrolled-AR coefficient matrix W in workspace.
// Layout: W[(kc*96 + t)*4 + kr]  with k = 4*kc + kr   (chunk-major so the
// GEMM kernel's B-fragment is one 8-byte load per lane).
// ---------------------------------------------------------------------------
__global__ void build_W_kernel(const float* __restrict__ phi,
                               const float* __restrict__ beta,
                               const float* __restrict__ constant,
                               float* __restrict__ W)
{
    __shared__ float h[PRED_T + 1];   // h[0..96]
    __shared__ float Hs[PRED_T];      // prefix sums of h

    if (threadIdx.x == 0) {
        const float p0 = phi[0], p1 = phi[1], p2 = phi[2];
        float hm1 = 0.f, hm2 = 0.f, hm3 = 0.f, run = 0.f;
        for (int t = 0; t <= PRED_T; ++t) {
            float ht = (t == 0) ? 1.0f : (p0 * hm1 + p1 * hm2 + p2 * hm3);
            h[t] = ht;
            hm3 = hm2; hm2 = hm1; hm1 = ht;
            if (t < PRED_T) { run += ht; Hs[t] = run; }
        }
    }
    __syncthreads();

    const float p1 = phi[1], p2 = phi[2];
    const float bt = beta[0];
    const float c0 = constant[0];

    for (int idx = threadIdx.x; idx < KDIM * PRED_T; idx += blockDim.x) {
        const int k = idx / PRED_T;
        const int t = idx - k * PRED_T;
        float w;
        if (k == 0)            w = p2 * h[t];                                   // coeff of L0
        else if (k == 1)       w = p1 * h[t] + ((t >= 1) ? p2 * h[t - 1] : 0.f);// coeff of L1
        else if (k == 2)       w = h[t + 1];                                    // coeff of L2
        else if (k == KDIM - 1) w = c0 * Hs[t];                                 // constant column
        else {
            const int s = k - 3;                                                // exog s
            w = (t >= s) ? bt * h[t - s] : 0.0f;
        }
        const int kc = k >> 2, kr = k & 3;
        W[(kc * PRED_T + t) * 4 + kr] = w;
    }
}

// ---------------------------------------------------------------------------
// Kernel 2: (65536 x 100) x (100 x 96) GEMM via V_WMMA_F32_16X16X4_F32.
// Block = 128 threads = 4 waves; each wave owns a 16-row M tile, six 16x16
// N tiles, looping K in 25 chunks of 4.
// ---------------------------------------------------------------------------
__global__ __launch_bounds__(128) void ar_wmma_kernel(
        const float* __restrict__ x_enc,   // (B, 512, 2)
        const float* __restrict__ W,       // chunk-major 100 x 96
        float* __restrict__ out)           // (B, 96)
{
    __shared__ float vbuf[ROWS_PER_BLOCK * VSTRIDE];

    const int tid       = threadIdx.x;
    const int blockBase = blockIdx.x * ROWS_PER_BLOCK;

    // ---- stage 64 rows of the K=100 input vector into LDS ----
    for (int idx = tid; idx < ROWS_PER_BLOCK * KDIM; idx += 128) {
        const int r = idx / KDIM;
        const int k = idx - r * KDIM;
        const int b = blockBase + r;
        float val;
        if (k < 3)              // lags: x_enc[b, 509+k, 0]
            val = x_enc[b * (SEQ_L * 2) + (509 + k) * 2 + 0];
        else if (k < 99)        // exog: x_enc[b, k-3, 1]
            val = x_enc[b * (SEQ_L * 2) + (k - 3) * 2 + 1];
        else
            val = 1.0f;         // ones column (constant term)
        vbuf[r * VSTRIDE + k] = val;
    }
    __syncthreads();

    const int wave    = tid >> 5;
    const int lane    = tid & 31;
    const int laneN   = lane & 15;
    const int halfSel = (lane < 16) ? 0 : 2;          // K sub-pair per half-wave
    const int rowLoc  = wave * 16 + laneN;            // A-matrix row for this lane

    v8f acc[6] = {};                                   // six 16x16 f32 tiles (N = 96)

    const float* vrow = &vbuf[rowLoc * VSTRIDE];
    for (int kc = 0; kc < KCHUNKS; ++kc) {
        // A fragment: 16x4 f32; lanes 0-15 hold K={0,1}, lanes 16-31 K={2,3}
        v2f a = *(const v2f*)(vrow + kc * 4 + halfSel);
#pragma unroll
        for (int n = 0; n < 6; ++n) {
            // B fragment: 4x16 f32, mirrored layout; single 8B load per lane
            v2f b = *(const v2f*)(W + (kc * PRED_T + n * 16 + laneN) * 4 + halfSel);
            acc[n] = __builtin_amdgcn_wmma_f32_16x16x4_f32(
                         /*neg_a=*/false, a, /*neg_b=*/false, b,
                         /*c_mod=*/(short)0, acc[n],
                         /*reuse_a=*/false, /*reuse_b=*/false);
        }
    }

    // ---- store D tiles: VGPR r -> M = r (lanes<16) / r+8 (lanes>=16), N = laneN
    const int mGlob = blockBase + wave * 16 + ((lane < 16) ? 0 : 8);
#pragma unroll
    for (int n = 0; n < 6; ++n) {
#pragma unroll
        for (int r = 0; r < 8; ++r) {
            out[(mGlob + r) * PRED_T + n * 16 + laneN] = acc[n][r];
        }
    }
}

// ---------------------------------------------------------------------------
extern "C" void kernel_launch(void* const* d_in, const int* in_sizes, int n_in,
                              void* d_out, int out_size, void* d_ws, size_t ws_size,
                              hipStream_t stream)
{
    // setup_inputs order: x_enc, x_mark_enc, x_dec, x_mark_dec, constant, phi, beta
    const float* x_enc    = (const float*)d_in[0];
    const float* constant = (const float*)d_in[4];
    const float* phi      = (const float*)d_in[5];
    const float* beta     = (const float*)d_in[6];
    float*       out      = (float*)d_out;
    float*       W        = (float*)d_ws;   // needs 100*96*4 = 38400 bytes

    build_W_kernel<<<1, 128, 0, stream>>>(phi, beta, constant, W);
    ar_wmma_kernel<<<B_TOTAL / ROWS_PER_BLOCK, 128, 0, stream>>>(x_enc, W, out);
}